// Elastic_50199577756037
// MI455X (gfx1250) — compile-verified
//
#include <hip/hip_runtime.h>
#include <math.h>

// ---------------- problem constants (match reference) ----------------
#define NYG   320
#define NXG   320
#define NTT   160
#define SHOTS 2
#define NSRC  8
#define NREC  96
#define PMLW  20
#define N2    (NYG*NXG)

#define DTc    4.0e-4f
#define INV_DH 0.25f
#define FC1    1.125f
#define FC2    ((float)(-1.0/24.0))

// ---------------- tiling ----------------
#define TS   32
#define HALO 2
#define LT   (TS + 2*HALO)   // 36 : velocity tile (owned + halo2)
#define ST   (TS + 4*2)      // 40 : stress tile (owned + halo4)

// ---------------- workspace layout (floats) ----------------
// ping-pong fields (read across block boundaries -> double buffered)
enum { P_VY=0, P_VX, P_SYY, P_SXY, P_SXX,
       P_MSYYY, P_MSXYY, P_MSXYX, P_MSXXX, NPP };
// in-place fields (owner-only read-modify-write)
enum { I_MVYY=0, I_MVYX, I_MVXY, I_MVXX, NIP };

#define PP_FLOATS (NPP*2*SHOTS*N2)
#define IP_OFF    PP_FLOATS
#define IP_FLOATS (NIP*SHOTS*N2)
#define B_OFF     (IP_OFF + IP_FLOATS)
#define REC_OFF   (B_OFF + NYG)
// recs_raw[t][s][r] at REC_OFF; total ws = REC_OFF + NTT*SHOTS*NREC floats (~18 MB)

// ---------------- gfx1250 async global->LDS (probe-verified present) ----------------
#ifndef HAVE_ASYNC_LDS
#if defined(__has_builtin)
#if __has_builtin(__builtin_amdgcn_global_load_async_to_lds_b32) && \
    __has_builtin(__builtin_amdgcn_s_wait_asynccnt)
#define HAVE_ASYNC_LDS 1
#endif
#endif
#endif
#ifndef HAVE_ASYNC_LDS
#define HAVE_ASYNC_LDS 0
#endif

__device__ __forceinline__ float* fpp(float* ws, int f, int p, int s) {
    return ws + (size_t)((f*2 + p)*SHOTS + s)*N2;
}
__device__ __forceinline__ float* fip(float* ws, int f, int s) {
    return ws + (size_t)IP_OFF + (size_t)(f*SHOTS + s)*N2;
}

__device__ __forceinline__ void lds_fill1(float* dst_lds, const float* src_glb) {
#if HAVE_ASYNC_LDS
    __builtin_amdgcn_global_load_async_to_lds_b32(
        (__attribute__((address_space(1))) int*)(src_glb),
        (__attribute__((address_space(3))) int*)(dst_lds),
        0, 0);
#else
    *dst_lds = *src_glb;
#endif
}

__device__ __forceinline__ void lds_fill_fence() {
#if HAVE_ASYNC_LDS
    __builtin_amdgcn_s_wait_asynccnt(0);
#endif
    __syncthreads();
}

// ---------------- init: zero state (both buffers), build PML decay b[] ----------------
__global__ void k_init(float* __restrict__ ws)
{
    unsigned i      = blockIdx.x*blockDim.x + threadIdx.x;
    unsigned stride = gridDim.x*blockDim.x;
    for (unsigned j = i; j < (unsigned)B_OFF; j += stride) ws[j] = 0.0f;
    if (i < NYG) {
        const double d0 = 3.0*1600.0*log(1000.0)/(2.0*(double)PMLW*4.0);
        double d = 0.0;
        int ii = (int)i;
        if (ii < PMLW) {
            double v = (double)(PMLW - ii)/(double)PMLW; d = d0*v*v;
        } else if (ii >= NYG - PMLW) {
            double v = (double)(ii - (NYG-1-PMLW))/(double)PMLW; d = d0*v*v;
        }
        ws[B_OFF + ii] = (float)exp(-d*4.0e-4);
    }
}

// ---------------- fused time step ----------------
// Phase A: velocities + ms* on extended tile (halo 2) from old-buffer stresses
//          (redundant halo compute; only owned 32x32 stored to new buffer)
// Phase B: stresses + mv* on owned tile from LDS velocities
__global__ __launch_bounds__(256)
void k_step(const float* __restrict__ lamb, const float* __restrict__ mu,
            const float* __restrict__ buoy,
            const float* __restrict__ src_amp,   // [SHOTS][NSRC][NTT]
            const int*   __restrict__ src_loc,   // [SHOTS][NSRC][2]
            const int*   __restrict__ rec_loc,   // [SHOTS][NREC][2]
            float* __restrict__ ws, int t, int pOld, int pNew)
{
    __shared__ float s_syy[ST*ST];
    __shared__ float s_sxy[ST*ST];
    __shared__ float s_sxx[ST*ST];
    __shared__ float s_vy[LT*LT];
    __shared__ float s_vx[LT*LT];

    const int s  = blockIdx.z;
    const int x0 = 2 + TS*blockIdx.x;
    const int y0 = 2 + TS*blockIdx.y;
    const int tid = threadIdx.y*32 + threadIdx.x;
    const float* b = ws + B_OFF;

    // ---- stress tile (old buffer) with halo 4 -> LDS ----
    const float* gsyyO = fpp(ws, P_SYY, pOld, s);
    const float* gsxyO = fpp(ws, P_SXY, pOld, s);
    const float* gsxxO = fpp(ws, P_SXX, pOld, s);
    for (int i = tid; i < ST*ST; i += 256) {
        int ly = i / ST, lx = i - ly*ST;
        int gy = y0 - 4 + ly, gx = x0 - 4 + lx;
        if (gy >= 0 && gy < NYG && gx >= 0 && gx < NXG) {
            int g = gy*NXG + gx;
            lds_fill1(&s_syy[i], gsyyO + g);
            lds_fill1(&s_sxy[i], gsxyO + g);
            lds_fill1(&s_sxx[i], gsxxO + g);
        } else {
            s_syy[i] = 0.0f; s_sxy[i] = 0.0f; s_sxx[i] = 0.0f;
        }
    }
    lds_fill_fence();

    // ---- Phase A: extended velocity update ----
    const float* gvyO = fpp(ws, P_VY, pOld, s);
    const float* gvxO = fpp(ws, P_VX, pOld, s);
    float*       gvyN = fpp(ws, P_VY, pNew, s);
    float*       gvxN = fpp(ws, P_VX, pNew, s);
    const float* gmsyyyO = fpp(ws, P_MSYYY, pOld, s);
    const float* gmsxyyO = fpp(ws, P_MSXYY, pOld, s);
    const float* gmsxyxO = fpp(ws, P_MSXYX, pOld, s);
    const float* gmsxxxO = fpp(ws, P_MSXXX, pOld, s);
    float* gmsyyyN = fpp(ws, P_MSYYY, pNew, s);
    float* gmsxyyN = fpp(ws, P_MSXYY, pNew, s);
    float* gmsxyxN = fpp(ws, P_MSXYX, pNew, s);
    float* gmsxxxN = fpp(ws, P_MSXXX, pNew, s);

    for (int i = tid; i < LT*LT; i += 256) {
        int ly = i / LT, lx = i - ly*LT;
        int y = y0 - HALO + ly, x = x0 - HALO + lx;  // y,x >= 0
        float vyv = 0.0f, vxv = 0.0f;
        if (y >= 2 && y < NYG-2 && x >= 2 && x < NXG-2) {
            const int sly = ly + 2, slx = lx + 2;    // into 40x40 stress tile
            const int g = y*NXG + x;
            const float byv = b[y];
            const float bxv = b[x];

            #define SYt(dy)     s_syy[(sly+(dy))*ST + slx]
            #define XYt(dy,dx)  s_sxy[(sly+(dy))*ST + slx + (dx)]
            #define SXt(dx)     s_sxx[(sly)*ST + slx + (dx)]

            // d/dy- syy
            float d1 = (FC1*(SYt(0) - SYt(-1)) + FC2*(SYt(1) - SYt(-2))) * INV_DH;
            float m1 = gmsyyyO[g];
            m1 = byv*m1 + (byv - 1.0f)*d1;
            float ay = d1 + m1;
            // d/dx+ sxy
            float d2 = (FC1*(XYt(0,1) - XYt(0,0)) + FC2*(XYt(0,2) - XYt(0,-1))) * INV_DH;
            float m2 = gmsxyxO[g];
            m2 = bxv*m2 + (bxv - 1.0f)*d2;
            ay += d2 + m2;

            const float bv = buoy[g];
            vyv = gvyO[g] + DTc*bv*ay;

            // source injection (accumulates duplicates like .at[].add)
            #pragma unroll
            for (int j = 0; j < NSRC; ++j) {
                int sy = src_loc[(s*NSRC + j)*2 + 0];
                int sx = src_loc[(s*NSRC + j)*2 + 1];
                if (sy == y && sx == x)
                    vyv += DTc * src_amp[(s*NSRC + j)*NTT + t] * bv;
            }

            // d/dx- sxx
            float d3 = (FC1*(SXt(0) - SXt(-1)) + FC2*(SXt(1) - SXt(-2))) * INV_DH;
            float m3 = gmsxxxO[g];
            m3 = bxv*m3 + (bxv - 1.0f)*d3;
            float ax = d3 + m3;
            // d/dy+ sxy
            float d4 = (FC1*(XYt(1,0) - XYt(0,0)) + FC2*(XYt(2,0) - XYt(-1,0))) * INV_DH;
            float m4 = gmsxyyO[g];
            m4 = byv*m4 + (byv - 1.0f)*d4;
            ax += d4 + m4;

            vxv = gvxO[g] + DTc*bv*ax;

            // store only the owned 32x32 interior
            if (ly >= HALO && ly < LT-HALO && lx >= HALO && lx < LT-HALO) {
                gvyN[g] = vyv;
                gvxN[g] = vxv;
                gmsyyyN[g] = m1;
                gmsxyxN[g] = m2;
                gmsxxxN[g] = m3;
                gmsxyyN[g] = m4;
            }
            #undef SYt
            #undef XYt
            #undef SXt
        }
        s_vy[i] = vyv;
        s_vx[i] = vxv;
    }
    __syncthreads();

    // ---- receiver record (vy is final for this step; owner block writes) ----
    if (tid < SHOTS*NREC) {
        int ss = tid / NREC, r = tid - ss*NREC;
        if (ss == s) {
            int ry = rec_loc[(ss*NREC + r)*2 + 0];
            int rx = rec_loc[(ss*NREC + r)*2 + 1];
            if (ry >= y0 && ry < y0+TS && rx >= x0 && rx < x0+TS &&
                ry < NYG-2 && rx < NXG-2)
                ws[REC_OFF + (t*SHOTS + ss)*NREC + r] =
                    s_vy[(ry - y0 + HALO)*LT + (rx - x0 + HALO)];
        }
    }

    // ---- Phase B: owned stress update from LDS velocities ----
    float* gsyyN = fpp(ws, P_SYY, pNew, s);
    float* gsxyN = fpp(ws, P_SXY, pNew, s);
    float* gsxxN = fpp(ws, P_SXX, pNew, s);
    float* gmvyy = fip(ws, I_MVYY, s);
    float* gmvyx = fip(ws, I_MVYX, s);
    float* gmvxy = fip(ws, I_MVXY, s);
    float* gmvxx = fip(ws, I_MVXX, s);

    const int x = x0 + threadIdx.x;
    if (x >= NXG-2) return;
    const float bxv = b[x];
    const int lx = x - x0 + HALO;

    #pragma unroll
    for (int k = 0; k < 4; ++k) {
        const int y = y0 + threadIdx.y + 8*k;
        if (y >= NYG-2) continue;
        const int ly = y - y0 + HALO;
        const int g  = y*NXG + x;
        const float byv = b[y];
        const int sidx = (y - y0 + 4)*ST + (x - x0 + 4);  // owned cell in stress tile

        #define VYl(dy,dx) s_vy[(ly+(dy))*LT + lx + (dx)]
        #define VXl(dy,dx) s_vx[(ly+(dy))*LT + lx + (dx)]

        // d/dy+ vy
        float e1d = (FC1*(VYl(1,0) - VYl(0,0)) + FC2*(VYl(2,0) - VYl(-1,0))) * INV_DH;
        float m = gmvyy[g];
        m = byv*m + (byv - 1.0f)*e1d;
        gmvyy[g] = m;
        float e1 = e1d + m;
        // d/dx- vx
        float e2d = (FC1*(VXl(0,0) - VXl(0,-1)) + FC2*(VXl(0,1) - VXl(0,-2))) * INV_DH;
        m = gmvxx[g];
        m = bxv*m + (bxv - 1.0f)*e2d;
        gmvxx[g] = m;
        float e2 = e2d + m;

        const float la  = lamb[g];
        const float muv = mu[g];
        const float l2m = la + 2.0f*muv;
        gsyyN[g] = s_syy[sidx] + DTc*(l2m*e1 + la*e2);
        gsxxN[g] = s_sxx[sidx] + DTc*(l2m*e2 + la*e1);

        // d/dx+ vy
        float gd = (FC1*(VYl(0,1) - VYl(0,0)) + FC2*(VYl(0,2) - VYl(0,-1))) * INV_DH;
        m = gmvyx[g];
        m = bxv*m + (bxv - 1.0f)*gd;
        gmvyx[g] = m;
        float gsum = gd + m;
        // d/dy- vx
        float gd2 = (FC1*(VXl(0,0) - VXl(-1,0)) + FC2*(VXl(1,0) - VXl(-2,0))) * INV_DH;
        m = gmvxy[g];
        m = byv*m + (byv - 1.0f)*gd2;
        gmvxy[g] = m;
        gsum += gd2 + m;

        gsxyN[g] = s_sxy[sidx] + DTc*muv*gsum;

        #undef VYl
        #undef VXl
    }
}

// ---------------- finalize: average adjacent time samples, transpose ----------------
__global__ void k_final(const float* __restrict__ ws, float* __restrict__ out)
{
    int i = blockIdx.x*blockDim.x + threadIdx.x;
    const int TOT = SHOTS*NREC*(NTT-1);
    if (i >= TOT) return;
    int tt = i % (NTT-1);
    int sr = i / (NTT-1);
    int r  = sr % NREC;
    int s  = sr / NREC;
    const float* rec = ws + REC_OFF;
    out[i] = 0.5f*(rec[((tt+1)*SHOTS + s)*NREC + r] + rec[(tt*SHOTS + s)*NREC + r]);
}

// ---------------- entry ----------------
extern "C" void kernel_launch(void* const* d_in, const int* in_sizes, int n_in,
                              void* d_out, int out_size, void* d_ws, size_t ws_size,
                              hipStream_t stream)
{
    const float* lamb = (const float*)d_in[0];
    const float* mu   = (const float*)d_in[1];
    const float* buoy = (const float*)d_in[2];
    const float* amps = (const float*)d_in[3];
    const int*   sloc = (const int*)d_in[4];
    const int*   rloc = (const int*)d_in[5];
    float* ws  = (float*)d_ws;
    float* out = (float*)d_out;

    k_init<<<4096, 256, 0, stream>>>(ws);

    dim3 blk(32, 8, 1);
    dim3 grd((NXG-4 + TS-1)/TS, (NYG-4 + TS-1)/TS, SHOTS);
    for (int t = 0; t < NTT; ++t) {
        int pOld = t & 1;
        k_step<<<grd, blk, 0, stream>>>(lamb, mu, buoy, amps, sloc, rloc,
                                        ws, t, pOld, pOld ^ 1);
    }

    const int TOT = SHOTS*NREC*(NTT-1);
    k_final<<<(TOT + 255)/256, 256, 0, stream>>>(ws, out);
}